// GNN_50663434224265
// MI455X (gfx1250) — compile-verified
//
#include <hip/hip_runtime.h>
#include <math.h>

typedef float v2f __attribute__((ext_vector_type(2)));
typedef float v8f __attribute__((ext_vector_type(8)));

// ---------------------------------------------------------------------------
// Exact-f32 WMMA GEMM: C[M,N] = f(A)[M,K] @ B[K,N] (+ cbias), using
// V_WMMA_F32_16X16X4_F32. One block = 16 rows of C, one wave per 16-col tile.
// f(A) = relu(A + abias) optionally (folds the previous layer's bias/ReLU
// into the A-staging pass). LDS row stride K+4 => stride mod 64 words == 4,
// so the two-float reads of both half-waves cover all 64 banks (no conflicts).
// Requires M % 16 == 0 (N=50000 = 3125*16 here).
// ---------------------------------------------------------------------------
template <int K, int NWAVES>
__global__ __launch_bounds__(NWAVES * 32)
void gemm_wmma_f32(const float* __restrict__ A, const float* __restrict__ B,
                   const float* __restrict__ abias, int arelu,
                   const float* __restrict__ cbias, float* __restrict__ C)
{
    constexpr int N   = NWAVES * 16;
    constexpr int LDA = K + 4;
    __shared__ float As[16 * LDA];

    const int tid  = threadIdx.x;
    const int lane = tid & 31;
    const int wave = tid >> 5;
    const int row0 = blockIdx.x * 16;
    const int n0   = wave * 16;

    // Cooperative stage of the 16xK A tile (uniform trip count -> no divergence).
    for (int idx = tid; idx < 16 * K; idx += NWAVES * 32) {
        const int m = idx / K;
        const int k = idx % K;
        float t = A[(long)(row0 + m) * K + k];
        if (abias) t += abias[k];
        if (arelu) t = fmaxf(t, 0.0f);
        As[m * LDA + k] = t;
    }
    __syncthreads();

    const int m    = lane & 15;   // M index (A row / C col-within-tile)
    const int half = lane >> 4;   // K half-select

    v8f acc;
    const float cb = cbias ? cbias[n0 + m] : 0.0f;
#pragma unroll
    for (int v = 0; v < 8; ++v) acc[v] = cb;

#pragma unroll
    for (int k0 = 0; k0 < K; k0 += 4) {
        const int kA = k0 + 2 * half;
        v2f a, b;
        // A 16x4 layout: lane m (half h) holds A[m][k0+2h], A[m][k0+2h+1]
        a.x = As[m * LDA + kA];
        a.y = As[m * LDA + kA + 1];
        // B 4x16 layout (mirror): lane m (half h) holds B[k0+2h][n0+m], B[k0+2h+1][n0+m]
        b.x = B[(long)kA * N + n0 + m];
        b.y = B[(long)(kA + 1) * N + n0 + m];
        acc = __builtin_amdgcn_wmma_f32_16x16x4_f32(false, a, false, b,
                                                    (short)0, acc, false, false);
    }

    // C/D layout: acc[v] -> C[row0 + v + 8*half][n0 + m]
#pragma unroll
    for (int v = 0; v < 8; ++v)
        C[(long)(row0 + v + 8 * half) * N + n0 + m] = acc[v];
}

// ---------------------------------------------------------------------------
// Degree / normalization
// ---------------------------------------------------------------------------
__global__ __launch_bounds__(256) void fill_ones(float* __restrict__ deg, int n)
{
    const int i = blockIdx.x * 256 + threadIdx.x;
    if (i < n) deg[i] = 1.0f;   // self-loop contributes 1 to every node
}

__global__ __launch_bounds__(256) void deg_scatter(const long long* __restrict__ ei,
                                                   float* __restrict__ deg, int E)
{
    const int e = blockIdx.x * 256 + threadIdx.x;
    if (e < E) atomicAdd(&deg[ei[E + e]], 1.0f);   // dst row of edge_index
}

__global__ __launch_bounds__(256) void inv_sqrt_deg(const float* __restrict__ deg,
                                                    float* __restrict__ isd, int n)
{
    const int i = blockIdx.x * 256 + threadIdx.x;
    if (i < n) isd[i] = rsqrtf(deg[i]);            // deg >= 1 always
}

// ---------------------------------------------------------------------------
// Aggregation: init with the self-loop term (h[i] * isd[i]^2), then
// scatter-add edge messages with device-scope f32 atomics (L2-resident).
// ---------------------------------------------------------------------------
template <int LOGF>
__global__ __launch_bounds__(256)
void self_loop_init(const float* __restrict__ H, const float* __restrict__ isd,
                    float* __restrict__ out, int total)
{
    const int idx = blockIdx.x * 256 + threadIdx.x;
    if (idx < total) {
        const float s = isd[idx >> LOGF];
        out[idx] = H[idx] * s * s;
    }
}

template <int F>   // one wave per edge; each lane handles F/32 features
__global__ __launch_bounds__(256)
void edge_scatter(const long long* __restrict__ ei, const float* __restrict__ isd,
                  const float* __restrict__ H, float* __restrict__ out, int E)
{
    const int w    = (blockIdx.x * 256 + threadIdx.x) >> 5;
    const int lane = threadIdx.x & 31;
    if (w >= E) return;
    const long long s = ei[w];
    const long long d = ei[E + w];
    const float     c = isd[s] * isd[d];
    if constexpr (F == 128) {
        const float4 v = ((const float4*)(H + s * 128))[lane];
        float* o = out + d * 128 + lane * 4;
        atomicAdd(o + 0, v.x * c);
        atomicAdd(o + 1, v.y * c);
        atomicAdd(o + 2, v.z * c);
        atomicAdd(o + 3, v.w * c);
    } else {
        const float2 v = ((const float2*)(H + s * 64))[lane];
        float* o = out + d * 64 + lane * 2;
        atomicAdd(o + 0, v.x * c);
        atomicAdd(o + 1, v.y * c);
    }
}

// ---------------------------------------------------------------------------
// branch() is evaluated twice on identical inputs (dropout is identity), so
// x1 == x2 and concat([x1,x1]) @ lin_W == x1 @ (lin_W_top + lin_W_bot).
// ---------------------------------------------------------------------------
__global__ __launch_bounds__(256) void make_wsum(const float* __restrict__ linW,
                                                 float* __restrict__ wsum)
{
    const int idx = blockIdx.x * 256 + threadIdx.x;
    if (idx < 64 * 64) wsum[idx] = linW[idx] + linW[idx + 64 * 64];
}

// log_softmax over 64 classes: one wave per row, two elements per lane.
__global__ __launch_bounds__(256) void logsoftmax64(float* __restrict__ out, int n)
{
    const int row  = (blockIdx.x * 256 + threadIdx.x) >> 5;
    const int lane = threadIdx.x & 31;
    if (row >= n) return;
    float* p = out + (long)row * 64;
    const float v0 = p[lane];
    const float v1 = p[lane + 32];
    float m = fmaxf(v0, v1);
#pragma unroll
    for (int o = 16; o > 0; o >>= 1) m = fmaxf(m, __shfl_xor(m, o, 32));
    float s = expf(v0 - m) + expf(v1 - m);
#pragma unroll
    for (int o = 16; o > 0; o >>= 1) s += __shfl_xor(s, o, 32);
    const float lse = m + logf(s);
    p[lane]      = v0 - lse;
    p[lane + 32] = v1 - lse;
}

// ---------------------------------------------------------------------------
// Orchestration
// ---------------------------------------------------------------------------
extern "C" void kernel_launch(void* const* d_in, const int* in_sizes, int n_in,
                              void* d_out, int out_size, void* d_ws, size_t ws_size,
                              hipStream_t stream)
{
    const float*     x    = (const float*)d_in[0];      // [n,128]
    const long long* ei   = (const long long*)d_in[1];  // [2,E] int64
    const float*     W1   = (const float*)d_in[2];      // [128,128]
    const float*     b1   = (const float*)d_in[3];      // [128]
    const float*     W2   = (const float*)d_in[4];      // [128,64]
    const float*     b2   = (const float*)d_in[5];      // [64]
    const float*     linW = (const float*)d_in[6];      // [128,64]
    const float*     linb = (const float*)d_in[7];      // [64]
    float*           out  = (float*)d_out;              // [n,64]

    const int n = in_sizes[0] / 128;   // 50000 (multiple of 16)
    const int E = in_sizes[1] / 2;     // 800000

    // Workspace carving (all buffers fully rewritten every call).
    const size_t bigBytes = (size_t)n * 128 * sizeof(float);          // 25.6 MB
    const size_t nBytes   = (((size_t)n * sizeof(float)) + 255) & ~(size_t)255;
    char*  w    = (char*)d_ws;
    float* buf0 = (float*)w;                       // h  (then reused for h2)
    float* buf1 = (float*)(w + bigBytes);          // agg1/h1 (then agg2/x1)
    float* deg  = (float*)(w + 2 * bigBytes);
    float* isd  = (float*)(w + 2 * bigBytes + nBytes);
    float* wsum = (float*)(w + 2 * bigBytes + 2 * nBytes);

    const int nbN = (n + 255) / 256;

    // Normalization coefficients
    fill_ones   <<<nbN, 256, 0, stream>>>(deg, n);
    deg_scatter <<<(E + 255) / 256, 256, 0, stream>>>(ei, deg, E);
    inv_sqrt_deg<<<nbN, 256, 0, stream>>>(deg, isd, n);
    make_wsum   <<<16, 256, 0, stream>>>(linW, wsum);

    // Layer 1: h = x @ W1  (WMMA), aggregate, (b1+ReLU folded into next GEMM)
    gemm_wmma_f32<128, 8><<<n / 16, 256, 0, stream>>>(x, W1, nullptr, 0, nullptr, buf0);
    self_loop_init<7><<<(n * 128 + 255) / 256, 256, 0, stream>>>(buf0, isd, buf1, n * 128);
    edge_scatter<128><<<(E + 7) / 8, 256, 0, stream>>>(ei, isd, buf0, buf1, E);

    // Layer 2: h2 = relu(agg1 + b1) @ W2, aggregate, (b2 folded into next GEMM)
    gemm_wmma_f32<128, 4><<<n / 16, 128, 0, stream>>>(buf1, W2, b1, 1, nullptr, buf0);
    self_loop_init<6><<<(n * 64 + 255) / 256, 256, 0, stream>>>(buf0, isd, buf1, n * 64);
    edge_scatter<64><<<(E + 7) / 8, 256, 0, stream>>>(ei, isd, buf0, buf1, E);

    // Fusion linear: out = (agg2 + b2) @ (linW_top + linW_bot) + lin_b
    gemm_wmma_f32<64, 4><<<n / 16, 128, 0, stream>>>(buf1, wsum, b2, 0, linb, out);

    // log_softmax rows of 64
    logsoftmax64<<<(n + 7) / 8, 256, 0, stream>>>(out, n);
}